// PointNet2_36172214567332
// MI455X (gfx1250) — compile-verified
//
#include <hip/hip_runtime.h>
#include <hip/hip_bf16.h>

typedef float v2f __attribute__((ext_vector_type(2)));
typedef float v8f __attribute__((ext_vector_type(8)));

#define TPB 256
#define BN_EPS 1e-5f

static __device__ __forceinline__ int imin(int a, int b) { return a < b ? a : b; }

// ---------------------------------------------------------------------------
// GEMM: Y[M,N] = X[M,K] @ W[N,K]^T + bias[N]   (one wave32 per 16x16 tile)
// Uses V_WMMA_F32_16X16X4_F32. EXEC must be all ones across the WMMA, so all
// out-of-range handling is done via clamped indices + mask multiplies.
// ---------------------------------------------------------------------------
__global__ void pn2_gemm_bias(const float* __restrict__ X, const float* __restrict__ W,
                              const float* __restrict__ bias, float* __restrict__ Y,
                              int M, int K, int N) {
  int tileM = blockIdx.x << 4;
  int tileN = blockIdx.y << 4;
  int lane  = threadIdx.x;        // 0..31
  int half  = lane >> 4;          // 0 or 1
  int l     = lane & 15;

  int row   = tileM + l;          // A-matrix row handled by this lane
  int rowC  = imin(row, M - 1);
  float rmask = (row < M) ? 1.f : 0.f;
  int col   = tileN + l;          // B-matrix column handled by this lane
  int colC  = imin(col, N - 1);
  float cmask = (col < N) ? 1.f : 0.f;

  v8f acc = {};
  for (int k0 = 0; k0 < K; k0 += 4) {
    int ka  = k0 + 2 * half;                 // VGPR0 K index for this lane half
    int ka0 = imin(ka,     K - 1);
    int ka1 = imin(ka + 1, K - 1);
    float km0 = (ka     < K) ? 1.f : 0.f;
    float km1 = (ka + 1 < K) ? 1.f : 0.f;
    v2f a, b;
    a.x = X[(size_t)rowC * K + ka0] * (rmask * km0);
    a.y = X[(size_t)rowC * K + ka1] * (rmask * km1);
    b.x = W[(size_t)colC * K + ka0] * (cmask * km0);
    b.y = W[(size_t)colC * K + ka1] * (cmask * km1);
    acc = __builtin_amdgcn_wmma_f32_16x16x4_f32(false, a, false, b, (short)0, acc,
                                                false, false);
  }
  // C/D layout: VGPR r -> M=tileM+r (lanes 0-15) / M=tileM+r+8 (lanes 16-31), N=tileN+l
  int outN = tileN + l;
  if (outN < N) {
    float bv = bias ? bias[outN] : 0.f;
    #pragma unroll
    for (int r = 0; r < 8; ++r) {
      int outM = tileM + r + 8 * half;
      if (outM < M) Y[(size_t)outM * N + outN] = acc[r] + bv;
    }
  }
}

// ---------------------------------------------------------------------------
// BatchNorm (train mode): per-channel mean / population variance over M rows.
// One block per channel.
// ---------------------------------------------------------------------------
__global__ void pn2_bn_stats(const float* __restrict__ X, int M, int C,
                             float* __restrict__ mean, float* __restrict__ var) {
  int c = blockIdx.x;
  __shared__ float s1[TPB], s2[TPB];
  float sum = 0.f, sq = 0.f;
  for (int m = threadIdx.x; m < M; m += TPB) {
    float v = X[(size_t)m * C + c];
    sum += v; sq += v * v;
  }
  s1[threadIdx.x] = sum; s2[threadIdx.x] = sq;
  __syncthreads();
  for (int s = TPB / 2; s > 0; s >>= 1) {
    if (threadIdx.x < s) { s1[threadIdx.x] += s1[threadIdx.x + s]; s2[threadIdx.x] += s2[threadIdx.x + s]; }
    __syncthreads();
  }
  if (threadIdx.x == 0) {
    float mu = s1[0] / (float)M;
    mean[c] = mu;
    var[c]  = s2[0] / (float)M - mu * mu;
  }
}

__global__ void pn2_bn_relu(float* __restrict__ X, const float* __restrict__ mean,
                            const float* __restrict__ var, const float* __restrict__ g,
                            const float* __restrict__ be, int total, int C) {
  int i = blockIdx.x * blockDim.x + threadIdx.x;
  if (i >= total) return;
  int c = i % C;
  float y = g[c] * (X[i] - mean[c]) * rsqrtf(var[c] + BN_EPS) + be[c];
  X[i] = fmaxf(y, 0.f);
}

// ---------------------------------------------------------------------------
// Input transpose: [B,3,N] -> [B,N,3]
// ---------------------------------------------------------------------------
__global__ void pn2_transpose_xyz(const float* __restrict__ in, float* __restrict__ out,
                                  int B, int N) {
  int i = blockIdx.x * blockDim.x + threadIdx.x;
  if (i >= B * N) return;
  int b = i / N, n = i % N;
  out[(size_t)i * 3 + 0] = in[((size_t)b * 3 + 0) * N + n];
  out[(size_t)i * 3 + 1] = in[((size_t)b * 3 + 1) * N + n];
  out[(size_t)i * 3 + 2] = in[((size_t)b * 3 + 2) * N + n];
}

// ---------------------------------------------------------------------------
// Farthest point sampling. One block per batch; argmax with first-index ties.
// ---------------------------------------------------------------------------
__global__ void pn2_fps(const float* __restrict__ xyz, int N, int npoint,
                        int* __restrict__ out_idx, float* __restrict__ dist_ws) {
  int b = blockIdx.x;
  const float* p = xyz + (size_t)b * N * 3;
  float* dist = dist_ws + (size_t)b * N;
  int* idx = out_idx + (size_t)b * npoint;
  __shared__ float s_val[TPB];
  __shared__ int   s_idx[TPB];
  int t = threadIdx.x;
  for (int i = t; i < N; i += TPB) dist[i] = 1e10f;
  __syncthreads();
  int farthest = 0;
  for (int it = 0; it < npoint; ++it) {
    if (t == 0) idx[it] = farthest;
    float cx = p[farthest * 3 + 0];
    float cy = p[farthest * 3 + 1];
    float cz = p[farthest * 3 + 2];
    float best = -1.f; int bestI = 0x7fffffff;
    for (int i = t; i < N; i += TPB) {
      float dx = p[i * 3 + 0] - cx;
      float dy = p[i * 3 + 1] - cy;
      float dz = p[i * 3 + 2] - cz;
      float d  = dx * dx + dy * dy + dz * dz;
      float dm = fminf(dist[i], d);
      dist[i] = dm;
      if (dm > best) { best = dm; bestI = i; }
    }
    s_val[t] = best; s_idx[t] = bestI;
    __syncthreads();
    for (int s = TPB / 2; s > 0; s >>= 1) {
      if (t < s) {
        if (s_val[t + s] > s_val[t] ||
            (s_val[t + s] == s_val[t] && s_idx[t + s] < s_idx[t])) {
          s_val[t] = s_val[t + s]; s_idx[t] = s_idx[t + s];
        }
      }
      __syncthreads();
    }
    farthest = s_idx[0];
    __syncthreads();
  }
}

// new_xyz = xyz[fps_idx]
__global__ void pn2_gather_xyz(const float* __restrict__ xyz, const int* __restrict__ idx,
                               int N, int S, int BS, float* __restrict__ out) {
  int i = blockIdx.x * blockDim.x + threadIdx.x;
  if (i >= BS) return;
  int b = i / S;
  int gi = idx[i];
  const float* p = xyz + ((size_t)b * N + gi) * 3;
  out[(size_t)i * 3 + 0] = p[0];
  out[(size_t)i * 3 + 1] = p[1];
  out[(size_t)i * 3 + 2] = p[2];
}

// ---------------------------------------------------------------------------
// Ball query: first K in-radius neighbors in index order, pad with first hit.
// ---------------------------------------------------------------------------
__global__ void pn2_ball_query(const float* __restrict__ xyz, const float* __restrict__ nxyz,
                               int N, int S, int K, float r2, int BS,
                               int* __restrict__ gidx) {
  int i = blockIdx.x * blockDim.x + threadIdx.x;
  if (i >= BS) return;
  int b = i / S;
  const float* p = xyz + (size_t)b * N * 3;
  const float* q = nxyz + (size_t)i * 3;
  float qx = q[0], qy = q[1], qz = q[2];
  int* out = gidx + (size_t)i * K;
  int cnt = 0, first = 0;
  for (int n = 0; n < N && cnt < K; ++n) {
    float dx = p[n * 3 + 0] - qx;
    float dy = p[n * 3 + 1] - qy;
    float dz = p[n * 3 + 2] - qz;
    if (dx * dx + dy * dy + dz * dz <= r2) {
      if (cnt == 0) first = n;
      out[cnt++] = n;
    }
  }
  for (; cnt < K; ++cnt) out[cnt] = first;
}

// feat[i, :] = [ points[b, gi, 0:D], xyz[b, gi]-new_xyz[b,s] ]
__global__ void pn2_group(const float* __restrict__ xyz, const float* __restrict__ nxyz,
                          const float* __restrict__ pts, int D, const int* __restrict__ gidx,
                          int N, int S, int K, int BSK, float* __restrict__ feat) {
  int i = blockIdx.x * blockDim.x + threadIdx.x;
  if (i >= BSK) return;
  int bs = i / K;
  int b  = bs / S;
  int gi = gidx[i];
  int C  = D + 3;
  float* o = feat + (size_t)i * C;
  if (pts) {
    const float* src = pts + ((size_t)b * N + gi) * D;
    for (int d = 0; d < D; ++d) o[d] = src[d];
  }
  const float* px = xyz + ((size_t)b * N + gi) * 3;
  const float* c  = nxyz + (size_t)bs * 3;
  o[D + 0] = px[0] - c[0];
  o[D + 1] = px[1] - c[1];
  o[D + 2] = px[2] - c[2];
}

// max over K neighbors, write into concatenated output at channel offset
__global__ void pn2_maxpool(const float* __restrict__ X, int K, int C,
                            float* __restrict__ out, int Ctot, int coff, int BS) {
  int i = blockIdx.x * blockDim.x + threadIdx.x;
  if (i >= BS * C) return;
  int r = i / C, c = i % C;
  const float* p = X + ((size_t)r * K) * C + c;
  float m = p[0];
  for (int k = 1; k < K; ++k) m = fmaxf(m, p[(size_t)k * C]);
  out[(size_t)r * Ctot + coff + c] = m;
}

// ---------------------------------------------------------------------------
// 3-NN with inverse-distance weights
// ---------------------------------------------------------------------------
__global__ void pn2_knn3(const float* __restrict__ xyz1, const float* __restrict__ xyz2,
                         int N1, int N2, int rows,
                         int* __restrict__ idx3, float* __restrict__ w3) {
  int i = blockIdx.x * blockDim.x + threadIdx.x;
  if (i >= rows) return;
  int b = i / N1;
  const float* q = xyz1 + (size_t)i * 3;
  const float* p = xyz2 + (size_t)b * N2 * 3;
  float qx = q[0], qy = q[1], qz = q[2];
  float d0 = 3.4e38f, d1 = 3.4e38f, d2 = 3.4e38f;
  int i0 = 0, i1 = 0, i2 = 0;
  for (int n = 0; n < N2; ++n) {
    float dx = p[n * 3 + 0] - qx;
    float dy = p[n * 3 + 1] - qy;
    float dz = p[n * 3 + 2] - qz;
    float d  = dx * dx + dy * dy + dz * dz;
    if (d < d0)      { d2 = d1; i2 = i1; d1 = d0; i1 = i0; d0 = d; i0 = n; }
    else if (d < d1) { d2 = d1; i2 = i1; d1 = d; i1 = n; }
    else if (d < d2) { d2 = d; i2 = n; }
  }
  float w0 = 1.f / (d0 + 1e-8f);
  float w1 = 1.f / (d1 + 1e-8f);
  float w2 = 1.f / (d2 + 1e-8f);
  float ws = w0 + w1 + w2;
  idx3[(size_t)i * 3 + 0] = i0; idx3[(size_t)i * 3 + 1] = i1; idx3[(size_t)i * 3 + 2] = i2;
  w3[(size_t)i * 3 + 0] = w0 / ws; w3[(size_t)i * 3 + 1] = w1 / ws; w3[(size_t)i * 3 + 2] = w2 / ws;
}

// dst[row, off+c] = sum_j w[row,j] * pts2[b, idx[row,j], c]
__global__ void pn2_interp3(const float* __restrict__ pts2, const int* __restrict__ idx3,
                            const float* __restrict__ w3, int N1, int N2, int C2,
                            float* __restrict__ dst, int Ctot, int off, int rows) {
  int i = blockIdx.x * blockDim.x + threadIdx.x;
  if (i >= rows * C2) return;
  int r = i / C2, c = i % C2;
  int b = r / N1;
  const float* pb = pts2 + (size_t)b * N2 * C2;
  float acc = 0.f;
  #pragma unroll
  for (int j = 0; j < 3; ++j) {
    int id = idx3[(size_t)r * 3 + j];
    acc += w3[(size_t)r * 3 + j] * pb[(size_t)id * C2 + c];
  }
  dst[(size_t)r * Ctot + off + c] = acc;
}

// dst[row, off+c] = src[row, c]
__global__ void pn2_copy_rows(const float* __restrict__ src, int Cs,
                              float* __restrict__ dst, int Ctot, int off, int rows) {
  int i = blockIdx.x * blockDim.x + threadIdx.x;
  if (i >= rows * Cs) return;
  int r = i / Cs, c = i % Cs;
  dst[(size_t)r * Ctot + off + c] = src[(size_t)r * Cs + c];
}

__global__ void pn2_logsoftmax(const float* __restrict__ X, float* __restrict__ out,
                               int M, int C) {
  int i = blockIdx.x * blockDim.x + threadIdx.x;
  if (i >= M) return;
  const float* x = X + (size_t)i * C;
  float m = x[0];
  for (int c = 1; c < C; ++c) m = fmaxf(m, x[c]);
  float s = 0.f;
  for (int c = 0; c < C; ++c) s += __expf(x[c] - m);
  float l = m + __logf(s);
  for (int c = 0; c < C; ++c) out[(size_t)i * C + c] = x[c] - l;
}

// ===========================================================================
// Host orchestration
// ===========================================================================
namespace {

struct ParamIt {
  void* const* d;
  int i;
  const float* next() { return (const float*)d[i++]; }
};

inline int cdiv(int a, int b) { return (a + b - 1) / b; }

// MLP (1x1 conv) + train-mode BN + ReLU, ping-pong between bufA/bufB.
// Input activations must be in bufIn. Returns pointer to final activations.
float* run_mlp(ParamIt& it, int nlayers, const int* widths, int cin,
               float* bufIn, float* bufA, float* bufB, int M,
               float* mean, float* var, hipStream_t stream) {
  float* in = bufIn;
  for (int l = 0; l < nlayers; ++l) {
    const float* W  = it.next();
    const float* b  = it.next();
    const float* g  = it.next();
    const float* be = it.next();
    int cout = widths[l];
    float* out = (in == bufA) ? bufB : bufA;
    dim3 grid(cdiv(M, 16), cdiv(cout, 16));
    pn2_gemm_bias<<<grid, 32, 0, stream>>>(in, W, b, out, M, cin, cout);
    pn2_bn_stats<<<cout, TPB, 0, stream>>>(out, M, cout, mean, var);
    int tot = M * cout;
    pn2_bn_relu<<<cdiv(tot, TPB), TPB, 0, stream>>>(out, mean, var, g, be, tot, cout);
    in = out; cin = cout;
  }
  return in;
}

} // namespace

extern "C" void kernel_launch(void* const* d_in, const int* in_sizes, int n_in,
                              void* d_out, int out_size, void* d_ws, size_t ws_size,
                              hipStream_t stream) {
  (void)in_sizes; (void)n_in; (void)out_size; (void)ws_size;
  const int B = 8, N0 = 8192;

  // ---- network configuration (mirrors SA_CFG / FP_CFG) ----
  static const int   SA_Nin[4]     = {8192, 1024, 256, 64};
  static const int   SA_S[4]       = {1024, 256, 64, 16};
  static const int   SA_D[4]       = {0, 96, 256, 512};
  static const float SA_R[4][2]    = {{0.05f, 0.1f}, {0.1f, 0.2f}, {0.2f, 0.4f}, {0.4f, 0.8f}};
  static const int   SA_K[4][2]    = {{16, 32}, {16, 32}, {16, 32}, {16, 32}};
  static const int   SA_W[4][2][3] = {{{16, 16, 32}, {32, 32, 64}},
                                      {{64, 64, 128}, {64, 96, 128}},
                                      {{128, 196, 256}, {128, 196, 256}},
                                      {{256, 256, 512}, {256, 384, 512}}};
  static const int   SA_Ctot[4]    = {96, 256, 512, 1024};
  static const int   FP_CIN[4]     = {1536, 512, 352, 128};
  static const int   FP_W[4][3]    = {{256, 256, 0}, {256, 256, 0}, {256, 128, 0}, {128, 128, 128}};
  static const int   FP_NL[4]      = {2, 2, 2, 3};

  // ---- workspace bump allocator ----
  size_t off = 0;
  auto alloc = [&](size_t elems, size_t esz) -> void* {
    off = (off + 255) & ~(size_t)255;
    void* p = (char*)d_ws + off;
    off += elems * esz;
    return p;
  };

  float* xyzT    = (float*)alloc((size_t)B * N0 * 3, 4);
  float* xs[5];
  xs[0] = xyzT;
  for (int l = 0; l < 4; ++l) xs[l + 1] = (float*)alloc((size_t)B * SA_S[l] * 3, 4);
  float* fs[5];
  fs[0] = nullptr;
  for (int l = 0; l < 4; ++l) fs[l + 1] = (float*)alloc((size_t)B * SA_S[l] * SA_Ctot[l], 4);
  float* bufA    = (float*)alloc((size_t)16777216, 4);  // max 262144 rows x 64 ch
  float* bufB    = (float*)alloc((size_t)16777216, 4);
  float* fps_dst = (float*)alloc((size_t)B * N0, 4);
  int*   fps_idx = (int*)  alloc((size_t)B * 1024, 4);
  int*   gidx    = (int*)  alloc((size_t)B * 1024 * 32, 4);
  int*   idx3    = (int*)  alloc((size_t)B * N0 * 3, 4);
  float* w3      = (float*)alloc((size_t)B * N0 * 3, 4);
  float* mean    = (float*)alloc(1024, 4);
  float* var     = (float*)alloc(1024, 4);
  float* fpres[4]; // fp outputs: fs3',fs2',fs1',f0
  fpres[0] = (float*)alloc((size_t)B * 64 * 256, 4);
  fpres[1] = (float*)alloc((size_t)B * 256 * 256, 4);
  fpres[2] = (float*)alloc((size_t)B * 1024 * 128, 4);
  fpres[3] = (float*)alloc((size_t)B * N0 * 128, 4);
  float* logits  = (float*)alloc((size_t)B * N0 * 13, 4);

  // Parameters consumed depth-first in dict-insertion order:
  // d_in[0]=xyz; then sa[level][branch][layer]=(W,b,g,be); then fp; then head.
  ParamIt it{d_in, 1};
  const float* xyz_in = (const float*)d_in[0];

  // ---- input transpose [B,3,N] -> [B,N,3] ----
  pn2_transpose_xyz<<<cdiv(B * N0, TPB), TPB, 0, stream>>>(xyz_in, xyzT, B, N0);

  // ---- set abstraction (MSG) levels ----
  for (int l = 0; l < 4; ++l) {
    int Nin = SA_Nin[l], S = SA_S[l], D = SA_D[l];
    int BS = B * S;
    pn2_fps<<<B, TPB, 0, stream>>>(xs[l], Nin, S, fps_idx, fps_dst);
    pn2_gather_xyz<<<cdiv(BS, TPB), TPB, 0, stream>>>(xs[l], fps_idx, Nin, S, BS, xs[l + 1]);
    int coff = 0;
    for (int br = 0; br < 2; ++br) {
      int K = SA_K[l][br];
      float r = SA_R[l][br];
      int BSK = BS * K;
      pn2_ball_query<<<cdiv(BS, TPB), TPB, 0, stream>>>(xs[l], xs[l + 1], Nin, S, K,
                                                        r * r, BS, gidx);
      pn2_group<<<cdiv(BSK, TPB), TPB, 0, stream>>>(xs[l], xs[l + 1], fs[l], D, gidx,
                                                    Nin, S, K, BSK, bufA);
      float* res = run_mlp(it, 3, SA_W[l][br], D + 3, bufA, bufA, bufB, BSK,
                           mean, var, stream);
      int Cbr = SA_W[l][br][2];
      pn2_maxpool<<<cdiv(BS * Cbr, TPB), TPB, 0, stream>>>(res, K, Cbr, fs[l + 1],
                                                           SA_Ctot[l], coff, BS);
      coff += Cbr;
    }
  }

  // ---- feature propagation levels (fp4, fp3, fp2, fp1) ----
  // (xyz1,N1,pts1,C1) <- (xyz2,N2,pts2,C2)
  const float* fp_x1[4] = {xs[3], xs[2], xs[1], xs[0]};
  const float* fp_x2[4] = {xs[4], xs[3], xs[2], xs[1]};
  const float* fp_p1[4] = {fs[3], fs[2], fs[1], nullptr};
  const float* fp_p2[4] = {fs[4], fpres[0], fpres[1], fpres[2]};
  const int    fp_N1[4] = {64, 256, 1024, N0};
  const int    fp_N2[4] = {16, 64, 256, 1024};
  const int    fp_C1[4] = {512, 256, 96, 0};
  const int    fp_C2[4] = {1024, 256, 256, 128};

  for (int l = 0; l < 4; ++l) {
    int N1 = fp_N1[l], N2 = fp_N2[l], C1 = fp_C1[l], C2 = fp_C2[l];
    int rows = B * N1;
    int Ctot = C1 + C2;
    pn2_knn3<<<cdiv(rows, TPB), TPB, 0, stream>>>(fp_x1[l], fp_x2[l], N1, N2, rows,
                                                  idx3, w3);
    if (fp_p1[l])
      pn2_copy_rows<<<cdiv(rows * C1, TPB), TPB, 0, stream>>>(fp_p1[l], C1, bufA,
                                                              Ctot, 0, rows);
    pn2_interp3<<<cdiv(rows * C2, TPB), TPB, 0, stream>>>(fp_p2[l], idx3, w3, N1, N2,
                                                          C2, bufA, Ctot, C1, rows);
    float* res = run_mlp(it, FP_NL[l], FP_W[l], Ctot, bufA, bufA, bufB, rows,
                         mean, var, stream);
    int Cout = FP_W[l][FP_NL[l] - 1];
    pn2_copy_rows<<<cdiv(rows * Cout, TPB), TPB, 0, stream>>>(res, Cout, fpres[l],
                                                              Cout, 0, rows);
  }

  // ---- head: linear128 + BN + ReLU -> linear13 -> log_softmax ----
  int M = B * N0;
  static const int headw[1] = {128};
  float* h = run_mlp(it, 1, headw, 128, fpres[3], bufA, bufB, M, mean, var, stream);
  const float* W2 = it.next();
  const float* b2 = it.next();
  {
    dim3 grid(cdiv(M, 16), cdiv(13, 16));
    pn2_gemm_bias<<<grid, 32, 0, stream>>>(h, W2, b2, logits, M, 128, 13);
  }
  pn2_logsoftmax<<<cdiv(M, TPB), TPB, 0, stream>>>(logits, (float*)d_out, M, 13);
}